// GAEModel_7404523618630
// MI455X (gfx1250) — compile-verified
//
#include <hip/hip_runtime.h>
#include <hip/hip_bf16.h>
#include <math.h>

typedef __attribute__((ext_vector_type(2))) float v2f;
typedef __attribute__((ext_vector_type(8))) float v8f;

#define NODE_DIM   64
#define HIDDEN_DIM 128
#define N_GRAPHS   8
#define NPG        2048
#define N_TOTAL    (N_GRAPHS * NPG)        // 16384
#define N_EDGES    (N_TOTAL * 32)          // 524288

// ---------------------------------------------------------------------------
// Zero the scatter accumulators (agg, deg, agg2) — must run every call since
// the harness does not re-poison between replays.
__global__ void zero3_kernel(float* __restrict__ a, float* __restrict__ b,
                             float* __restrict__ c) {
    int n = blockIdx.x * blockDim.x + threadIdx.x;
    if (n < N_TOTAL) { a[n] = 0.f; b[n] = 0.f; c[n] = 0.f; }
}

// ---------------------------------------------------------------------------
// s[n] = sum_k X[n,k]   (K = 64 or 128, rows 16B aligned -> float4 loads)
__global__ void rowsum_kernel(const float* __restrict__ X, float* __restrict__ s,
                              int K) {
    int n = blockIdx.x * blockDim.x + threadIdx.x;
    if (n >= N_TOTAL) return;
    const float4* p = (const float4*)(X + (size_t)n * K);
    float acc = 0.f;
    int nv = K >> 2;
    for (int i = 0; i < nv; ++i) {
        float4 v = p[i];
        acc += v.x + v.y + v.z + v.w;
    }
    s[n] = acc;
}

// ---------------------------------------------------------------------------
// Edge scatter: agg[dst] += s[src]; (optionally) deg[dst] += 1.
// edge_index is int64: [0:E) = src, [E:2E) = dst.
__global__ void edge_scatter_kernel(const long long* __restrict__ ei,
                                    const float* __restrict__ s,
                                    float* __restrict__ agg,
                                    float* __restrict__ deg) {
    int e = blockIdx.x * blockDim.x + threadIdx.x;
    if (e >= N_EDGES) return;
    int src = (int)ei[e];
    int dst = (int)ei[(size_t)N_EDGES + e];
    atomicAdd(&agg[dst], s[src]);
    if (deg) atomicAdd(&deg[dst], 1.0f);
}

// agg[n] /= max(deg[n], 1)   (mean aggregation; isolated nodes -> 0)
__global__ void normalize_kernel(float* __restrict__ agg,
                                 const float* __restrict__ deg) {
    int n = blockIdx.x * blockDim.x + threadIdx.x;
    if (n >= N_TOTAL) return;
    float d = deg[n];
    agg[n] = agg[n] / (d > 0.f ? d : 1.f);
}

// ---------------------------------------------------------------------------
// out[row,col] = relu( aggn[row] + sum_k A[row,k]*B[k,col] + bias[col] )
// A: [N_TOTAL, K] row-major, B: [K, 128] row-major, out: [N_TOTAL, 128].
// One wave32 computes one 16x16 output tile via V_WMMA_F32_16X16X4_F32
// (exact f32 matrix path, matches reference precision). 4 waves / block.
// Grid covers all tiles exactly -> EXEC all-ones at every WMMA.
__global__ void nnconv_gemm_wmma_kernel(const float* __restrict__ A,
                                        const float* __restrict__ B,
                                        const float* __restrict__ aggn,
                                        const float* __restrict__ bias,
                                        float* __restrict__ out, int K) {
    const int wave = threadIdx.x >> 5;
    const int lane = threadIdx.x & 31;
    const int tile = blockIdx.x * (blockDim.x >> 5) + wave;
    const int mt = tile >> 3;            // 128/16 = 8 N-tiles
    const int nt = tile & 7;
    const int M0 = mt << 4;
    const int N0 = nt << 4;
    const int half = lane >> 4;          // 0: K pair {0,1}, 1: K pair {2,3}
    const int lm   = lane & 15;

    v8f c = {};                          // 16x16 f32 accumulator (8 VGPRs)
    // A-frag source: row (M0+lm), starting at K = 2*half, step 4.
    const float* arow = A + (size_t)(M0 + lm) * K + 2 * half;
    // B-frag source: col (N0+lm), rows (k + 2*half) and (k + 2*half + 1).
    const float* bcol = B + (size_t)(2 * half) * HIDDEN_DIM + N0 + lm;

    for (int k = 0; k < K; k += 4) {
        v2f a = *(const v2f*)(arow + k);                 // 8B-aligned
        v2f b;
        const float* bp = bcol + (size_t)k * HIDDEN_DIM;
        b.x = bp[0];
        b.y = bp[HIDDEN_DIM];
        // D = A(16x4) * B(4x16) + C   -> v_wmma_f32_16x16x4_f32
        c = __builtin_amdgcn_wmma_f32_16x16x4_f32(
                /*neg_a=*/false, a, /*neg_b=*/false, b,
                /*c_mod=*/(short)0, c, /*reuse_a=*/false, /*reuse_b=*/false);
    }

    const int col = N0 + lm;
    const float bc = bias[col];
#pragma unroll
    for (int j = 0; j < 8; ++j) {
        int row = M0 + j + 8 * half;     // C/D layout: VGPR j, lane-half selects M+8
        float v = c[j] + aggn[row] + bc;
        out[(size_t)row * HIDDEN_DIM + col] = fmaxf(v, 0.f);
    }
}

// ---------------------------------------------------------------------------
// left[n] = h[n,:] . w_lin[0:128] ; right[n] = h[n,:] . w_lin[128:256]
__global__ void lr_kernel(const float* __restrict__ h,
                          const float* __restrict__ w_lin,
                          float* __restrict__ left, float* __restrict__ right) {
    int n = blockIdx.x * blockDim.x + threadIdx.x;
    if (n >= N_TOTAL) return;
    const float4* hp = (const float4*)(h + (size_t)n * HIDDEN_DIM);
    const float4* wl = (const float4*)(w_lin);
    const float4* wr = (const float4*)(w_lin + HIDDEN_DIM);
    float l = 0.f, r = 0.f;
#pragma unroll 8
    for (int i = 0; i < HIDDEN_DIM / 4; ++i) {
        float4 hv = hp[i];
        float4 a = wl[i], b = wr[i];
        l += hv.x * a.x + hv.y * a.y + hv.z * a.z + hv.w * a.w;
        r += hv.x * b.x + hv.y * b.y + hv.z * b.z + hv.w * b.w;
    }
    left[n] = l;
    right[n] = r;
}

// ---------------------------------------------------------------------------
// scores[b, i, j] = sigmoid(left[b,i] + right[b,j] + b_lin)
// 33.5M outputs -> bandwidth bound; 4 outputs (one float4 store) per thread.
__global__ void scores_kernel(const float* __restrict__ left,
                              const float* __restrict__ right,
                              const float* __restrict__ b_lin,
                              float* __restrict__ out) {
    const size_t NN = (size_t)NPG * NPG;
    size_t idx = ((size_t)blockIdx.x * blockDim.x + threadIdx.x) * 4;
    int b = (int)(idx / NN);
    size_t rem = idx - (size_t)b * NN;
    int i = (int)(rem >> 11);            // / 2048
    int j = (int)(rem & (NPG - 1));      // 2048-aligned rows, j multiple of 4
    float l = left[b * NPG + i] + b_lin[0];
    float4 r = *(const float4*)(right + b * NPG + j);
    float4 o;
    o.x = 1.f / (1.f + __expf(-(l + r.x)));
    o.y = 1.f / (1.f + __expf(-(l + r.y)));
    o.z = 1.f / (1.f + __expf(-(l + r.z)));
    o.w = 1.f / (1.f + __expf(-(l + r.w)));
    *(float4*)(out + idx) = o;
}

// ---------------------------------------------------------------------------
extern "C" void kernel_launch(void* const* d_in, const int* in_sizes, int n_in,
                              void* d_out, int out_size, void* d_ws, size_t ws_size,
                              hipStream_t stream) {
    (void)in_sizes; (void)n_in; (void)out_size; (void)ws_size;
    const float*     x     = (const float*)d_in[0];
    const long long* ei    = (const long long*)d_in[1];   // int64 edge_index
    const float*     W1    = (const float*)d_in[2];
    const float*     b1    = (const float*)d_in[3];
    const float*     W2    = (const float*)d_in[4];
    const float*     b2    = (const float*)d_in[5];
    const float*     w_lin = (const float*)d_in[6];
    const float*     b_lin = (const float*)d_in[7];

    // Small scratch in d_ws (7 * 64 KB = 448 KB).
    float* ws   = (float*)d_ws;
    float* s1   = ws;                    // [N_TOTAL]
    float* agg  = ws + 1 * N_TOTAL;
    float* deg  = ws + 2 * N_TOTAL;
    float* s2   = ws + 3 * N_TOTAL;
    float* agg2 = ws + 4 * N_TOTAL;
    float* left = ws + 5 * N_TOTAL;
    float* right= ws + 6 * N_TOTAL;

    // Large intermediates live inside d_out (128 MB); fully consumed before
    // the final scores kernel overwrites all of d_out.
    float* outf = (float*)d_out;
    float* h1 = outf;                          // 2M floats = 8 MB
    float* h2 = outf + (size_t)N_TOTAL * HIDDEN_DIM;

    const int T = 256;
    const int nodeBlocks = N_TOTAL / T;        // 64
    const int edgeBlocks = N_EDGES / T;        // 2048
    const int gemmBlocks = (N_TOTAL / 16) * (HIDDEN_DIM / 16) / 4;  // 2048 (4 waves/blk)

    zero3_kernel<<<nodeBlocks, T, 0, stream>>>(agg, deg, agg2);

    // ---- Conv layer 1 ----
    rowsum_kernel<<<nodeBlocks, T, 0, stream>>>(x, s1, NODE_DIM);
    edge_scatter_kernel<<<edgeBlocks, T, 0, stream>>>(ei, s1, agg, deg);
    normalize_kernel<<<nodeBlocks, T, 0, stream>>>(agg, deg);
    nnconv_gemm_wmma_kernel<<<gemmBlocks, 128, 0, stream>>>(x, W1, agg, b1, h1, NODE_DIM);

    // ---- Conv layer 2 (deg is edge-only -> reuse) ----
    rowsum_kernel<<<nodeBlocks, T, 0, stream>>>(h1, s2, HIDDEN_DIM);
    edge_scatter_kernel<<<edgeBlocks, T, 0, stream>>>(ei, s2, agg2, nullptr);
    normalize_kernel<<<nodeBlocks, T, 0, stream>>>(agg2, deg);
    nnconv_gemm_wmma_kernel<<<gemmBlocks, 128, 0, stream>>>(h1, W2, agg2, b2, h2, HIDDEN_DIM);

    // ---- Link predictor ----
    lr_kernel<<<nodeBlocks, T, 0, stream>>>(h2, w_lin, left, right);

    const size_t nOut = (size_t)N_GRAPHS * NPG * NPG;      // 33,554,432
    const int scoreBlocks = (int)(nOut / 4 / T);           // 32768
    scores_kernel<<<scoreBlocks, T, 0, stream>>>(left, right, b_lin, outf);
}